// AttentionLayerRouter_51943334477902
// MI455X (gfx1250) — compile-verified
//
#include <hip/hip_runtime.h>
#include <hip/hip_bf16.h>
#include <stdint.h>

// ---------------------------------------------------------------------------
// AttentionLayerRouter, batch-0-only single pass over text_features[0].
//   K1 (128 WGs): stage 16x4096 f32 tile in 320KB LDS (async-to-LDS),
//       WMMA f32 16x16x4 MLP -> scores, chunk-local softmax partials,
//       weighted partial pool from LDS.
//   K2 (1 WG): log-sum-exp merge of 128 chunks, L2-normalize * sqrt(4096).
//   K3 (1 WG): router MLP via WMMA (pooled broadcast to all 16 A-rows),
//       exact GELU, logits, softmax(T=2), top-8, renormalize, write 40 outs.
// ---------------------------------------------------------------------------

#define DIM     4096
#define HID     256
#define NLAYER  24
#define TOPK    8
#define LSEQ    2048
#define CHUNK   16
#define NCHUNK  (LSEQ / CHUNK)    // 128
#define XSTRIDE 4100              // padded f32 row stride for X tile in LDS
#define WSTRIDE 36                // padded f32 row stride for W slab in LDS
#define KSLAB   32

typedef float v2f __attribute__((ext_vector_type(2)));
typedef float v8f __attribute__((ext_vector_type(8)));
typedef int   v4i __attribute__((vector_size(16)));   // matches builtin proto

#if defined(__has_builtin)
#  if __has_builtin(__builtin_amdgcn_global_load_async_to_lds_b128) && \
      __has_builtin(__builtin_amdgcn_s_wait_asynccnt)
#    define USE_ASYNC_LDS 1
#  endif
#endif

typedef __attribute__((address_space(1))) v4i* g4p;   // global int4-vec ptr
typedef __attribute__((address_space(3))) v4i* l4p;   // LDS int4-vec ptr

__device__ __forceinline__ void copy16_to_lds(const void* gsrc, void* ldst) {
#ifdef USE_ASYNC_LDS
  __builtin_amdgcn_global_load_async_to_lds_b128((g4p)gsrc, (l4p)ldst, 0, 0);
#else
  *(float4*)ldst = *(const float4*)gsrc;
#endif
}

__device__ __forceinline__ void async_wait_all() {
#ifdef USE_ASYNC_LDS
  __builtin_amdgcn_s_wait_asynccnt(0);
#endif
}

__device__ __forceinline__ v8f wmma_f32x4(v2f a, v2f b, v8f c) {
  return __builtin_amdgcn_wmma_f32_16x16x4_f32(false, a, false, b,
                                               (short)0, c, false, false);
}

__device__ __forceinline__ float rsum16(float v) {
  v += __shfl_xor(v, 1, 16);
  v += __shfl_xor(v, 2, 16);
  v += __shfl_xor(v, 4, 16);
  v += __shfl_xor(v, 8, 16);
  return v;
}

// ---------------------------------------------------------------------------
// Kernel 1: per 16-token chunk -> scores, chunk softmax stats, weighted psum
// ---------------------------------------------------------------------------
__global__ __launch_bounds__(256) void k1_scores_psum(
    const float* __restrict__ X, const float* __restrict__ W1,
    const float* __restrict__ b1, const float* __restrict__ W2,
    const float* __restrict__ b2, float* __restrict__ psum,
    float* __restrict__ stats) {
  extern __shared__ float lds[];
  float* Xs   = lds;                    // 16 * 4100
  float* Ws   = Xs + CHUNK * XSTRIDE;   // 256 * 36
  float* scrS = Ws + HID * WSTRIDE;     // 16 tiles x 16 rows
  float* wts  = scrS + 256;             // 16
  float* sArr = wts + 16;               // 16

  const int tid   = threadIdx.x;
  const int lane  = tid & 31;
  const int l16   = lane & 15;
  const int hi    = lane >> 4;          // K-half select
  const int khalf = hi * 2;
  const int wv    = tid >> 5;
  const int chunk = blockIdx.x;

  // ---- stage X[chunk*16 .. +16, :] into LDS (async, padded rows) ----
  {
    const char* gb = (const char*)(X + (size_t)chunk * CHUNK * DIM);
    for (int q = tid; q < CHUNK * (DIM / 4); q += 256) {   // 16B granules
      const int row = q >> 10, c = q & 1023;
      copy16_to_lds(gb + (size_t)row * (DIM * 4) + (size_t)c * 16,
                    (char*)Xs + ((size_t)row * XSTRIDE + c * 4) * 4);
    }
  }

  const int   t0 = 2 * wv, t1 = t0 + 1;          // two 16-wide N tiles / wave
  const int   n0 = t0 * 16 + l16, n1 = t1 * 16 + l16;
  const float bias0 = b1[n0], bias1 = b1[n1];
  const float w2v0 = W2[n0], w2v1 = W2[n1];

  v8f acc0 = {};
  v8f acc1 = {};

  for (int kb = 0; kb < DIM; kb += KSLAB) {
    __syncthreads();  // previous W slab fully consumed
    for (int q = tid; q < HID * (KSLAB / 4); q += 256) {
      const int row = q >> 3, c = q & 7;
      copy16_to_lds((const char*)W1 + ((size_t)row * DIM + kb) * 4 + c * 16,
                    (char*)Ws + ((size_t)row * WSTRIDE + c * 4) * 4);
    }
    async_wait_all();   // also covers the X tile on the first iteration
    __syncthreads();

#pragma unroll
    for (int kk = 0; kk < KSLAB; kk += 4) {
      const int ka = kb + kk + khalf;
      v2f a   = *(const v2f*)(Xs + (size_t)l16 * XSTRIDE + ka);
      v2f bb0 = *(const v2f*)(Ws + (size_t)n0 * WSTRIDE + kk + khalf);
      v2f bb1 = *(const v2f*)(Ws + (size_t)n1 * WSTRIDE + kk + khalf);
      acc0 = wmma_f32x4(a, bb0, acc0);
      acc1 = wmma_f32x4(a, bb1, acc1);
    }
  }

  // ---- scores: s[m] = b2 + sum_n relu(H[m,n]+b1[n]) * W2[n] ----
#pragma unroll
  for (int i = 0; i < 8; ++i) {
    float h0 = fmaxf(acc0[i] + bias0, 0.0f);
    float h1 = fmaxf(acc1[i] + bias1, 0.0f);
    float r0 = rsum16(h0 * w2v0);
    float r1 = rsum16(h1 * w2v1);
    if (l16 == 0) {                       // lanes 0 and 16: rows i / i+8
      scrS[t0 * 16 + i + hi * 8] = r0;
      scrS[t1 * 16 + i + hi * 8] = r1;
    }
  }
  __syncthreads();
  if (tid < CHUNK) {
    float s = b2[0];
    for (int t = 0; t < 16; ++t) s += scrS[t * 16 + tid];
    sArr[tid] = s;
  }
  __syncthreads();
  if (tid < CHUNK) {
    float mx = sArr[0];
    for (int i = 1; i < CHUNK; ++i) mx = fmaxf(mx, sArr[i]);
    wts[tid] = expf(sArr[tid] - mx);
  }
  __syncthreads();
  if (tid == 0) {
    float mx = sArr[0];
    for (int i = 1; i < CHUNK; ++i) mx = fmaxf(mx, sArr[i]);
    float S = 0.f;
    for (int i = 0; i < CHUNK; ++i) S += wts[i];
    stats[2 * chunk]     = mx;
    stats[2 * chunk + 1] = S;
  }

  // ---- weighted partial pool straight out of LDS (X read once from HBM) ----
  for (int d = tid; d < DIM; d += 256) {
    float acc = 0.f;
#pragma unroll
    for (int l = 0; l < CHUNK; ++l) acc += wts[l] * Xs[(size_t)l * XSTRIDE + d];
    psum[(size_t)chunk * DIM + d] = acc;
  }
}

// ---------------------------------------------------------------------------
// Kernel 2: merge 128 chunks (log-sum-exp), normalize, scale by sqrt(DIM)
// ---------------------------------------------------------------------------
__global__ __launch_bounds__(1024) void k2_pool(
    const float* __restrict__ psum, const float* __restrict__ stats,
    float* __restrict__ pooledN) {
  __shared__ float coef[NCHUNK];
  __shared__ float red[1024];
  __shared__ float sc[2];
  const int tid = threadIdx.x;

  if (tid == 0) {
    float M = stats[0];
    for (int c = 1; c < NCHUNK; ++c) M = fmaxf(M, stats[2 * c]);
    sc[0] = M;
  }
  __syncthreads();
  if (tid < NCHUNK) coef[tid] = expf(stats[2 * tid] - sc[0]);
  __syncthreads();
  if (tid == 0) {
    float T = 0.f;
    for (int c = 0; c < NCHUNK; ++c) T += stats[2 * c + 1] * coef[c];
    sc[1] = T;
  }

  float pr[DIM / 1024];
  float ssq = 0.f;
#pragma unroll
  for (int i = 0; i < DIM / 1024; ++i) {
    const int d = tid + i * 1024;
    float a = 0.f;
    for (int c = 0; c < NCHUNK; ++c) a += coef[c] * psum[(size_t)c * DIM + d];
    pr[i] = a;
    ssq += a * a;
  }
  red[tid] = ssq;
  __syncthreads();
  if (tid == 0) {
    float s = 0.f;
    for (int i = 0; i < 1024; ++i) s += red[i];
    // pooled = praw/T; normalize: praw * 64 / max(||praw||, 1e-12*T)
    sc[0] = 64.0f / fmaxf(sqrtf(s), 1e-12f * sc[1]);
  }
  __syncthreads();
#pragma unroll
  for (int i = 0; i < DIM / 1024; ++i)
    pooledN[tid + i * 1024] = pr[i] * sc[0];
}

// ---------------------------------------------------------------------------
// Kernel 3: router MLP (WMMA, pooled broadcast to all 16 A rows) + top-k
// ---------------------------------------------------------------------------
__global__ __launch_bounds__(256) void k3_router(
    const float* __restrict__ pooledN, const float* __restrict__ R1,
    const float* __restrict__ Rb1, const float* __restrict__ R2,
    const float* __restrict__ Rb2, float* __restrict__ out) {
  __shared__ float pn[DIM];
  __shared__ float Ws[HID * WSTRIDE];
  __shared__ float rbuf[HID];
  __shared__ float lg[NLAYER];

  const int tid   = threadIdx.x;
  const int lane  = tid & 31;
  const int l16   = lane & 15;
  const int khalf = (lane >> 4) * 2;
  const int wv    = tid >> 5;

  for (int d = tid; d < DIM; d += 256) pn[d] = pooledN[d];

  const int t0 = 2 * wv, t1 = t0 + 1;
  const int n0 = t0 * 16 + l16, n1 = t1 * 16 + l16;

  v8f acc0 = {};
  v8f acc1 = {};

  for (int kb = 0; kb < DIM; kb += KSLAB) {
    __syncthreads();
    for (int q = tid; q < HID * (KSLAB / 4); q += 256) {
      const int row = q >> 3, c = q & 7;
      copy16_to_lds((const char*)R1 + ((size_t)row * DIM + kb) * 4 + c * 16,
                    (char*)Ws + ((size_t)row * WSTRIDE + c * 4) * 4);
    }
    async_wait_all();
    __syncthreads();

#pragma unroll
    for (int kk = 0; kk < KSLAB; kk += 4) {
      const int ka = kb + kk + khalf;
      v2f a;                       // same vector in every A row (broadcast)
      a.x = pn[ka];
      a.y = pn[ka + 1];
      v2f bb0 = *(const v2f*)(Ws + (size_t)n0 * WSTRIDE + kk + khalf);
      v2f bb1 = *(const v2f*)(Ws + (size_t)n1 * WSTRIDE + kk + khalf);
      acc0 = wmma_f32x4(a, bb0, acc0);
      acc1 = wmma_f32x4(a, bb1, acc1);
    }
  }

  // row 0 of C lives in accN[0] on lanes 0..15
  if (lane < 16) {
    float x0 = acc0[0] + Rb1[n0];
    rbuf[n0] = 0.5f * x0 * (1.0f + erff(x0 * 0.70710678118654752f));
    float x1 = acc1[0] + Rb1[n1];
    rbuf[n1] = 0.5f * x1 * (1.0f + erff(x1 * 0.70710678118654752f));
  }
  __syncthreads();

  if (tid < NLAYER) {
    float s = Rb2[tid];
    for (int h = 0; h < HID; ++h) s += rbuf[h] * R2[tid * HID + h];
    lg[tid] = s;
  }
  __syncthreads();

  if (tid == 0) {
    float mx = lg[0];
    for (int j = 1; j < NLAYER; ++j) mx = fmaxf(mx, lg[j]);
    float p[NLAYER];
    float S = 0.f;
    for (int j = 0; j < NLAYER; ++j) {
      p[j] = expf((lg[j] - mx) * 0.5f);   // temperature = 2.0
      S += p[j];
    }
    for (int j = 0; j < NLAYER; ++j) p[j] /= S;

    float q[NLAYER];
    for (int j = 0; j < NLAYER; ++j) q[j] = p[j];
    int   ti[TOPK];
    float tw[TOPK];
    float tsum = 0.f;
    for (int i = 0; i < TOPK; ++i) {      // strict '>' keeps lowest index on tie
      int bi = 0;
      float bv = q[0];
      for (int j = 1; j < NLAYER; ++j)
        if (q[j] > bv) { bv = q[j]; bi = j; }
      ti[i] = bi;
      tw[i] = bv;
      tsum += bv;
      q[bi] = -1.0f;
    }
    for (int i = 0; i < TOPK; ++i) {
      out[i]         = (float)ti[i];
      out[TOPK + i]  = tw[i] / tsum;
    }
    for (int j = 0; j < NLAYER; ++j) out[2 * TOPK + j] = p[j];
  }
}

// ---------------------------------------------------------------------------
extern "C" void kernel_launch(void* const* d_in, const int* in_sizes, int n_in,
                              void* d_out, int out_size, void* d_ws,
                              size_t ws_size, hipStream_t stream) {
  (void)in_sizes; (void)n_in; (void)out_size; (void)ws_size;
  const float* X   = (const float*)d_in[0];  // batch 0 = first 2048*4096 f32
  const float* W1  = (const float*)d_in[1];
  const float* b1  = (const float*)d_in[2];
  const float* W2  = (const float*)d_in[3];
  const float* b2  = (const float*)d_in[4];
  const float* R1  = (const float*)d_in[5];
  const float* Rb1 = (const float*)d_in[6];
  const float* R2  = (const float*)d_in[7];
  const float* Rb2 = (const float*)d_in[8];

  float* ws      = (float*)d_ws;
  float* psum    = ws;                         // 128*4096
  float* stats   = ws + (size_t)NCHUNK * DIM;  // 2*128
  float* pooledN = stats + 2 * NCHUNK;         // 4096

  const size_t lds1 =
      (size_t)(CHUNK * XSTRIDE + HID * WSTRIDE + 256 + 32) * sizeof(float);

  k1_scores_psum<<<NCHUNK, 256, lds1, stream>>>(X, W1, b1, W2, b2, psum, stats);
  k2_pool<<<1, 1024, 0, stream>>>(psum, stats, pooledN);
  k3_router<<<1, 256, 0, stream>>>(pooledN, R1, Rb1, R2, Rb2, (float*)d_out);
}